// QuantumLayer_90331752170355
// MI455X (gfx1250) — compile-verified
//
#include <hip/hip_runtime.h>
#include <hip/hip_bf16.h>
#include <math.h>
#include <stdint.h>

typedef float v2f __attribute__((ext_vector_type(2)));
typedef float v8f __attribute__((ext_vector_type(8)));

// ---------------------------------------------------------------------------
// Kernel 1: build A = Re(M^H Z M) (64x64, real symmetric) from the 54 params.
// One workgroup, runs once. Threads 0..63 each simulate the circuit on basis
// state |j> (their column of M, in LDS); columns are independent. Then all
// 256 threads form A and write it to d_ws.
// ---------------------------------------------------------------------------
__global__ __launch_bounds__(256) void build_A_kernel(const float* __restrict__ params,
                                                      float* __restrict__ Aout) {
    __shared__ float sRe[64][65];
    __shared__ float sIm[64][65];
    __shared__ float sp[54];

    const int tid = threadIdx.x;
    if (tid < 54) sp[tid] = params[tid];
    __syncthreads();

    if (tid < 64) {
        const int j = tid;
#pragma clang loop unroll(disable)
        for (int k = 0; k < 64; ++k) {
            sRe[k][j] = (k == j) ? 1.0f : 0.0f;
            sIm[k][j] = 0.0f;
        }

        // Apply complex 2x2 on qubit q of this column; ctrl >= 0 => controlled.
        auto apply = [&](int q, int ctrl,
                         float a00r, float a00i, float a01r, float a01i,
                         float a10r, float a10i, float a11r, float a11i) {
            const int bit = 1 << q;
#pragma clang loop unroll(disable)
            for (int m = 0; m < 32; ++m) {
                const int k0 = ((m >> q) << (q + 1)) | (m & (bit - 1));
                if (ctrl >= 0 && !((k0 >> ctrl) & 1)) continue;
                const int k1 = k0 | bit;
                const float ar = sRe[k0][j], ai = sIm[k0][j];
                const float br = sRe[k1][j], bi = sIm[k1][j];
                sRe[k0][j] = a00r * ar - a00i * ai + a01r * br - a01i * bi;
                sIm[k0][j] = a00r * ai + a00i * ar + a01r * bi + a01i * br;
                sRe[k1][j] = a10r * ar - a10i * ai + a11r * br - a11i * bi;
                sIm[k1][j] = a10r * ai + a10i * ar + a11r * bi + a11i * br;
            }
        };

        // qiskit U(theta,phi,lam) = [[c, -e^{i lam} s],[e^{i phi} s, e^{i(phi+lam)} c]]
        auto U3g = [&](int q, int ctrl, int p) {
            const float th = sp[p], ph = sp[p + 1], la = sp[p + 2];
            const float c = cosf(0.5f * th), s = sinf(0.5f * th);
            float sl, cl, sph, cph, spl, cpl;
            sincosf(la, &sl, &cl);
            sincosf(ph, &sph, &cph);
            sincosf(ph + la, &spl, &cpl);
            apply(q, ctrl,
                  c, 0.0f, -cl * s, -sl * s,
                  cph * s, sph * s, cpl * c, spl * c);
        };
        auto CX = [&](int qc, int qt) {
            apply(qt, qc, 0.f, 0.f, 1.f, 0.f, 1.f, 0.f, 0.f, 0.f);
        };
        auto Pg = [&](int q, int p) {
            float sl, cl;
            sincosf(sp[p], &sl, &cl);
            apply(q, -1, 1.f, 0.f, 0.f, 0.f, 0.f, 0.f, cl, sl);
        };

        // Layer 1: U on all qubits (params 0..17)
#pragma clang loop unroll(disable)
        for (int q = 0; q < 6; ++q) U3g(q, -1, 3 * q);
        // cx(2,1); p(18,1); cx(4,3); p(19,3)
        CX(2, 1); Pg(1, 18); CX(4, 3); Pg(3, 19);
        // Layer 2: U on {0,1,3,5} (params 20..31)
        U3g(0, -1, 20); U3g(1, -1, 23); U3g(3, -1, 26); U3g(5, -1, 29);
        // CU(32..34) ctrl0->tgt1 ; CU(35..37) ctrl3->tgt5
        U3g(1, 0, 32); U3g(5, 3, 35);
        // cx(3,1); p(38,1)
        CX(3, 1); Pg(1, 38);
        // Final: U(39) q0; U(42) q1; CU(45) ctrl0->tgt1; U(48) q0; U(51) q1
        U3g(0, -1, 39); U3g(1, -1, 42); U3g(1, 0, 45); U3g(0, -1, 48); U3g(1, -1, 51);
    }
    __syncthreads();

    // A[i][j] = sum_k z_k (Mr[k][i]*Mr[k][j] + Mi[k][i]*Mi[k][j]),  z_k = +-1
#pragma clang loop unroll(disable)
    for (int e = tid; e < 64 * 64; e += 256) {
        const int i = e >> 6, jc = e & 63;
        float acc = 0.0f;
#pragma unroll 4
        for (int k = 0; k < 64; ++k) {
            const float z = (k < 32) ? 1.0f : -1.0f;
            acc += z * (sRe[k][i] * sRe[k][jc] + sIm[k][i] * sIm[k][jc]);
        }
        Aout[e] = acc;
    }
}

// ---------------------------------------------------------------------------
// Kernel 2: out[b] = (f_b^T A f_b) / (f_b^T f_b), fallback A[0][0] if ||f||~0.
// 256 threads = 8 wave32; each wave processes ILOOP consecutive 16-row tiles,
// software-pipelined with two LDS buffers:
//   - F tiles: GLOBAL_LOAD_ASYNC_TO_LDS_B128 (ASYNCcnt), prefetch next tile
//     while WMMA-ing the current one (in-order completion => s_wait_asynccnt 8).
//   - A: staged once per WG, packed as row pairs -> B operand = one ds_load_b64.
//   - GEMM Y = F*A via v_wmma_f32_16x16x4_f32 (fp32, no precision loss):
//     4 N-tiles x 16 K-steps = 64 WMMAs per tile.
// ---------------------------------------------------------------------------
#define WAVES 8
#define ILOOP 4

__global__ __launch_bounds__(256) void qexp_kernel(const float* __restrict__ F,
                                                   const float* __restrict__ Amat,
                                                   float* __restrict__ out, int B) {
    // Apk[p][c] = (A[2p][c], A[2p+1][c]) : B-operand halves contiguous (b64).
    __shared__ __align__(16) v2f   Apk[32][66];
    __shared__ __align__(16) float Flds[WAVES][2][16][68]; // double-buffered F

    const int tid  = threadIdx.x;
    const int lane = tid & 31;
    const int w    = tid >> 5;
    const int half = lane >> 4;   // 0: lanes 0-15, 1: lanes 16-31
    const int l16  = lane & 15;

    const long ntiles = (long)B >> 4;                    // full 16-row tiles
    const long tile0  = ((long)blockIdx.x * WAVES + w) * ILOOP;

    // Issue one 16x64 tile (4 KB) as 8 per-lane async b128 copies into LDS.
    auto issue = [&](int buf, long tile) {
        const float* gbase = F + (size_t)(tile << 4) * 64; // wave-uniform -> SGPR
#pragma unroll
        for (int it = 0; it < 8; ++it) {
            const int flat = it * 128 + lane * 4;          // 16B per lane
            const int r = flat >> 6, c = flat & 63;
            const unsigned lds_off = (unsigned)(uintptr_t)&Flds[w][buf][r][c];
            const unsigned goff    = (unsigned)(flat * 4);
            asm volatile("global_load_async_to_lds_b128 %0, %1, %2"
                         :: "v"(lds_off), "v"(goff), "s"(gbase)
                         : "memory");
        }
    };

    if (tile0 < ntiles) issue(0, tile0);

    // Stage A into LDS, packed as row pairs (coalesced; L2-resident grid-wide).
    for (int e = tid; e < 64 * 64; e += 256) {
        const int r = e >> 6, c = e & 63;
        ((float*)&Apk[r >> 1][c])[r & 1] = Amat[e];
    }
    __syncthreads();

    const float A00 = ((const float*)&Apk[0][0])[0];

    for (int g = 0; g < ILOOP; ++g) {                    // wave-uniform loop
        const long tg = tile0 + g;
        if (tg >= ntiles) break;

        const bool have_next = (g + 1 < ILOOP) && (tg + 1 < ntiles);
        if (have_next) {
            // WAR guard: prior ds reads from the target buffer are done.
            asm volatile("s_wait_dscnt 0" ::: "memory");
            issue((g + 1) & 1, tg + 1);
            asm volatile("s_wait_asynccnt 8" ::: "memory");  // current tile landed
        } else {
            asm volatile("s_wait_asynccnt 0" ::: "memory");
        }

        const float (*Ft)[68] = Flds[w][g & 1];
        const long row0 = tg << 4;

        // A-operand (16x4 f32): lane L -> row M=L%16, K = 2*(L/16)+{0,1}.
        v2f aop[16];
#pragma unroll
        for (int k = 0; k < 16; ++k) {
            aop[k] = *(const v2f*)&Ft[l16][4 * k + 2 * half];
        }

        float qsum[8], nsum[8];
#pragma unroll
        for (int v = 0; v < 8; ++v) { qsum[v] = 0.0f; nsum[v] = 0.0f; }

#pragma unroll
        for (int n = 0; n < 4; ++n) {
            v8f acc = {};
#pragma unroll
            for (int k = 0; k < 16; ++k) {
                // B-operand (4x16 f32): VGPR0 rows K={0,2}, VGPR1 rows K={1,3},
                // N=l16. Row pair (4k+2*half, +1) == packed pair index 2k+half.
                const v2f b = Apk[2 * k + half][n * 16 + l16];
                acc = __builtin_amdgcn_wmma_f32_16x16x4_f32(
                    false, aop[k], false, b, (short)0, acc, false, false);
            }
            // C/D layout: lane = N + 16*(M>=8), VGPR v = M%8.
#pragma unroll
            for (int v = 0; v < 8; ++v) {
                const float fe = Ft[v + 8 * half][n * 16 + l16];
                qsum[v] += fe * acc[v];
                nsum[v] += fe * fe;
            }
        }

        // Reduce across the 16 lanes of each half-wave.
#pragma unroll
        for (int v = 0; v < 8; ++v) {
            float q = qsum[v], s = nsum[v];
#pragma unroll
            for (int off = 1; off < 16; off <<= 1) {
                q += __shfl_xor(q, off, 32);
                s += __shfl_xor(s, off, 32);
            }
            qsum[v] = q; nsum[v] = s;
        }

        if (l16 == 0) {
#pragma unroll
            for (int v = 0; v < 8; ++v) {
                const float s = nsum[v];
                // jnp.isclose(norm, 0) -> norm <= 1e-8 -> norm^2 <= 1e-16
                out[row0 + v + 8 * half] = (s <= 1e-16f) ? A00 : (qsum[v] / s);
            }
        }
    }
}

extern "C" void kernel_launch(void* const* d_in, const int* in_sizes, int n_in,
                              void* d_out, int out_size, void* d_ws, size_t ws_size,
                              hipStream_t stream) {
    const float* features = (const float*)d_in[0];
    const float* params   = (const float*)d_in[1];
    float* out = (float*)d_out;
    float* Aws = (float*)d_ws;   // 64*64 floats = 16 KB

    const int B = in_sizes[0] / 64;

    build_A_kernel<<<1, 256, 0, stream>>>(params, Aws);

    const int ntiles  = (B + 15) / 16;
    const int nblocks = (ntiles + WAVES * ILOOP - 1) / (WAVES * ILOOP);
    qexp_kernel<<<nblocks, 256, 0, stream>>>(features, Aws, out, B);
}